// GNNEncoder_90271622627497
// MI455X (gfx1250) — compile-verified
//
#include <hip/hip_runtime.h>
#include <hip/hip_bf16.h>

typedef __attribute__((ext_vector_type(2))) float v2f;
typedef __attribute__((ext_vector_type(8))) float v8f;

#define IN_DIM 256
#define HID    256
#define EMB    64
#define BN_EPS 1e-3f

// ---------------------------------------------------------------------------
// Dense GEMM  C[M,N] = A[M,K] * B[K,N]  (row-major, fp32) via V_WMMA_F32_16X16X4_F32
// One wave32 computes a 16x64 strip of C (4 accumulators of 16x16).
// Requires: M%16==0, K%4==0, N%64==0.
//
// fp32 WMMA operand layout (CDNA5 ISA 7.12.2):
//   A 16x4 : lanes 0-15 hold row=lane,   K={0,1} in the two VGPRs;
//            lanes 16-31 hold row=lane-16, K={2,3}.
//   B 4x16 : VGPR v, lanes 0-15  -> K=v,   N=lane
//            VGPR v, lanes 16-31 -> K=v+2, N=lane-16
//   C 16x16: VGPR r, lanes 0-15 -> M=r, N=lane; lanes 16-31 -> M=r+8, N=lane-16
// ---------------------------------------------------------------------------
__launch_bounds__(256)
__global__ void gemm_wmma_f32(const float* __restrict__ A,
                              const float* __restrict__ B,
                              float* __restrict__ C,
                              int M, int K, int N)
{
    const int lane    = threadIdx.x & 31;   // gfx1250 is wave32
    const int wave    = threadIdx.x >> 5;   // 8 waves per 256-thread block
    const int laneRow = lane & 15;
    const int hi      = lane >> 4;          // 0: lanes 0-15, 1: lanes 16-31
    const int kOff    = hi * 2;             // K offset for A/B fragments

    const int colGroups  = N >> 6;          // 64-wide column groups
    const int totalUnits = (M >> 4) * colGroups;
    const int unitStride = gridDim.x * 8;

    for (int unit = blockIdx.x * 8 + wave; unit < totalUnits; unit += unitStride) {
        const int row0 = (unit / colGroups) << 4;
        const int col0 = (unit % colGroups) << 6;

        v8f acc0 = {}, acc1 = {}, acc2 = {}, acc3 = {};

        const float* Arow = A + (size_t)(row0 + laneRow) * K + kOff;

        // Prefetch the next row tile's A data (speculative, wave-uniform branch).
        if (row0 + 16 < M)
            __builtin_prefetch(A + (size_t)(row0 + 16 + laneRow) * K, 0, 0);

        for (int k0 = 0; k0 < K; k0 += 4) {
            // A fragment: two consecutive fp32 (8B aligned: K even offsets)
            v2f a = *(const v2f*)(Arow + k0);

            // B fragments for the 4 column tiles of this 64-wide strip
            const float* Bk0 = B + (size_t)(k0 + kOff) * N + col0 + laneRow;
            const float* Bk1 = Bk0 + N;
            v2f b0 = { Bk0[0],  Bk1[0]  };
            v2f b1 = { Bk0[16], Bk1[16] };
            v2f b2 = { Bk0[32], Bk1[32] };
            v2f b3 = { Bk0[48], Bk1[48] };

            acc0 = __builtin_amdgcn_wmma_f32_16x16x4_f32(false, a, false, b0, (short)0, acc0, false, false);
            acc1 = __builtin_amdgcn_wmma_f32_16x16x4_f32(false, a, false, b1, (short)0, acc1, false, false);
            acc2 = __builtin_amdgcn_wmma_f32_16x16x4_f32(false, a, false, b2, (short)0, acc2, false, false);
            acc3 = __builtin_amdgcn_wmma_f32_16x16x4_f32(false, a, false, b3, (short)0, acc3, false, false);
        }

        // Store: VGPR r -> row (row0 + r + 8*hi), col (col0 + laneRow + 16*t)
        float* Crow = C + (size_t)(row0 + hi * 8) * N + col0 + laneRow;
        #pragma unroll
        for (int r = 0; r < 8; ++r) {
            float* cr = Crow + (size_t)r * N;
            cr[0]  = acc0[r];
            cr[16] = acc1[r];
            cr[32] = acc2[r];
            cr[48] = acc3[r];
        }
    }
}

// ---------------------------------------------------------------------------
// SpMM scatter: out[row_e, :] += val_e * dense[col_e, :]
// D/4 lanes per edge, float4 gather + 4 fp32 global atomics each.
// Gathered rows are L2-resident (dense matrix is ~102 MB < 192 MB L2).
// ---------------------------------------------------------------------------
template <int D>
__launch_bounds__(256)
__global__ void spmm_scatter(const int* __restrict__ rows,
                             const int* __restrict__ cols,
                             const float* __restrict__ vals,
                             const float* __restrict__ dense,
                             float* __restrict__ out,
                             long long E)
{
    constexpr int TPE = D / 4;      // threads per edge
    constexpr int EPB = 256 / TPE;  // edges per block per iteration
    const int t  = threadIdx.x % TPE;
    const int el = threadIdx.x / TPE;
    const int d0 = t * 4;

    for (long long e = (long long)blockIdx.x * EPB + el; e < E;
         e += (long long)gridDim.x * EPB) {
        const int   r = rows[e];
        const int   c = cols[e];
        const float v = vals[e];
        const float4 g = *(const float4*)(dense + (size_t)c * D + d0);
        float* o = out + (size_t)r * D + d0;
        atomicAdd(o + 0, v * g.x);
        atomicAdd(o + 1, v * g.y);
        atomicAdd(o + 2, v * g.z);
        atomicAdd(o + 3, v * g.w);
    }
}

// ---------------------------------------------------------------------------
// BatchNorm (inference) + optional ReLU, in place.  D must be a power of 2.
// ---------------------------------------------------------------------------
__launch_bounds__(256)
__global__ void bn_act(float* __restrict__ h,
                       const float* __restrict__ gamma,
                       const float* __restrict__ beta,
                       const float* __restrict__ mean,
                       const float* __restrict__ var,
                       long long total, int D, int relu)
{
    for (long long i = (long long)blockIdx.x * 256 + threadIdx.x; i < total;
         i += (long long)gridDim.x * 256) {
        const int col = (int)(i & (long long)(D - 1));
        float y = (h[i] - mean[col]) * rsqrtf(var[col] + BN_EPS) * gamma[col] + beta[col];
        if (relu) y = fmaxf(y, 0.0f);
        h[i] = y;
    }
}

// ---------------------------------------------------------------------------
extern "C" void kernel_launch(void* const* d_in, const int* in_sizes, int n_in,
                              void* d_out, int out_size, void* d_ws, size_t ws_size,
                              hipStream_t stream)
{
    const float* x      = (const float*)d_in[0];
    const int*   erow   = (const int*)  d_in[1];
    const int*   ecol   = (const int*)  d_in[2];
    const float* eval_  = (const float*)d_in[3];
    const float* w1     = (const float*)d_in[4];
    const float* w2     = (const float*)d_in[5];
    const float* gamma1 = (const float*)d_in[6];
    const float* beta1  = (const float*)d_in[7];
    const float* mean1  = (const float*)d_in[8];
    const float* var1   = (const float*)d_in[9];
    const float* gamma2 = (const float*)d_in[10];
    const float* beta2  = (const float*)d_in[11];
    const float* mean2  = (const float*)d_in[12];
    const float* var2   = (const float*)d_in[13];

    const int       N = in_sizes[0] / IN_DIM;   // 100000
    const long long E = in_sizes[1];            // 3.2M

    float* bufA = (float*)d_ws;                 // N*HID fp32 (h0 / later z0)
    float* bufB = bufA + (size_t)N * HID;       // N*HID fp32 (h accumulator)
    float* z    = (float*)d_out;                // N*EMB fp32

    // Zero accumulation targets (graph-capture safe).
    hipMemsetAsync(bufB, 0, (size_t)N * HID * sizeof(float), stream);
    hipMemsetAsync(z,    0, (size_t)N * EMB * sizeof(float), stream);

    // 1) h0 = x @ w1      [N,256]x[256,256]
    {
        int units  = (N / 16) * (HID / 64);
        int blocks = (units + 7) / 8;
        gemm_wmma_f32<<<blocks, 256, 0, stream>>>(x, w1, bufA, N, IN_DIM, HID);
    }
    // 2) h = A @ h0 (scatter-add over edges)
    {
        constexpr int EPB = 256 / (HID / 4);    // 4 edges / block
        long long nb = (E + EPB - 1) / EPB;
        int blocks = nb > 400000 ? 400000 : (int)nb;
        spmm_scatter<HID><<<blocks, 256, 0, stream>>>(erow, ecol, eval_, bufA, bufB, E);
    }
    // 3) h = relu(bn1(h)) in place
    {
        long long total = (long long)N * HID;
        long long nb = (total + 255) / 256;
        int blocks = nb > 100000 ? 100000 : (int)nb;
        bn_act<<<blocks, 256, 0, stream>>>(bufB, gamma1, beta1, mean1, var1, total, HID, 1);
    }
    // 4) z0 = h @ w2      [N,256]x[256,64]  (reuse bufA)
    {
        int units  = (N / 16) * (EMB / 64);
        int blocks = (units + 7) / 8;
        gemm_wmma_f32<<<blocks, 256, 0, stream>>>(bufB, w2, bufA, N, HID, EMB);
    }
    // 5) z = A @ z0 (scatter-add into d_out)
    {
        constexpr int EPB = 256 / (EMB / 4);    // 16 edges / block
        long long nb = (E + EPB - 1) / EPB;
        int blocks = nb > 400000 ? 400000 : (int)nb;
        spmm_scatter<EMB><<<blocks, 256, 0, stream>>>(erow, ecol, eval_, bufA, z, E);
    }
    // 6) z = bn2(z) in place (no relu)
    {
        long long total = (long long)N * EMB;
        long long nb = (total + 255) / 256;
        int blocks = nb > 100000 ? 100000 : (int)nb;
        bn_act<<<blocks, 256, 0, stream>>>(z, gamma2, beta2, mean2, var2, total, EMB, 0);
    }
}